// AgentTransformer_40424232190498
// MI455X (gfx1250) — compile-verified
//
#include <hip/hip_runtime.h>
#include <cmath>

#define T 61
#define MPAD 64   // padded M rows for WMMA tiles
#define L 60
#define DIM 128
#define NH 8
#define HD 16
#define AGENT 5
#define KSIZE 9
#define ASTRIDE 12
#define SCALEF 0.25f

#define XLD 132   // padded row stride for x tile
#define QLD 388   // padded row stride for qkv tile
#define OLD 132   // padded row stride for h / attn-out tile

typedef __attribute__((ext_vector_type(2))) float v2f;
typedef __attribute__((ext_vector_type(8))) float v8f;

__device__ __forceinline__ float wave_sum(float v) {
#pragma unroll
  for (int off = 16; off; off >>= 1) v += __shfl_xor(v, off, 32);
  return v;
}
__device__ __forceinline__ float wave_max(float v) {
#pragma unroll
  for (int off = 16; off; off >>= 1) v = fmaxf(v, __shfl_xor(v, off, 32));
  return v;
}

extern "C" __global__ void __launch_bounds__(256)
agent_attn_fused(const float* __restrict__ x,
                 const float* __restrict__ w_qkv,  const float* __restrict__ b_qkv,
                 const float* __restrict__ w_proj, const float* __restrict__ b_proj,
                 const float* __restrict__ w_projq,const float* __restrict__ b_projq,
                 const float* __restrict__ conv1_w,const float* __restrict__ conv1_b,
                 const float* __restrict__ ln_w,   const float* __restrict__ ln_b,
                 const float* __restrict__ conv2_w,
                 const float* __restrict__ an_bias,const float* __restrict__ na_bias,
                 const float* __restrict__ at_bias,const float* __restrict__ ta_bias,
                 float* __restrict__ out)
{
  __shared__ __align__(16) float sX[MPAD * XLD];   // x[b], rows 61..63 zero
  __shared__ __align__(16) float sQKV[MPAD * QLD]; // q|k|v, 64 x 384 (rows>=61 junk)
  __shared__ __align__(16) float sHO[MPAD * OLD];  // conv h rows 0..59, later attn-out rows 0..60; 61..63 zero
  __shared__ float sAT[NH * AGENT * T];            // bias_at, then attn1 in place
  __shared__ float sTA[NH * T * AGENT];            // bias_ta, then attn2 in place
  __shared__ float sAgentQ[NH * AGENT * HD];
  __shared__ float sAgentV[NH * AGENT * HD];
  __shared__ int   sPos[AGENT];

  const int b       = blockIdx.x;
  const int tid     = threadIdx.x;
  const int lane    = tid & 31;
  const int wv      = tid >> 5;
  const int lane_lo = lane & 15;
  const int lane_hi = lane >> 4;

  const float* xb = x + (size_t)b * (T * DIM);

  // ---- P0: stage x[b]; zero pad rows; build bias_at / bias_ta ----
  for (int e = tid; e < T * DIM; e += 256)
    sX[(e >> 7) * XLD + (e & 127)] = xb[e];
  for (int e = tid; e < (MPAD - T) * XLD; e += 256) sX[T * XLD + e] = 0.0f;
  for (int e = tid; e < (MPAD - T) * OLD; e += 256) sHO[T * OLD + e] = 0.0f;

  for (int e = tid; e < NH * AGENT * T; e += 256) {       // [h][a][t] flat == at_bias flat
    int t  = e % T;
    int ha = e / T;
    float coord = (t + 0.5f) * ((float)AGENT / (float)T) - 0.5f;
    coord = fminf(fmaxf(coord, 0.0f), (float)(AGENT - 1));
    int i0 = (int)floorf(coord);
    int i1 = min(i0 + 1, AGENT - 1);
    float w = coord - (float)i0;
    sAT[e] = an_bias[ha * AGENT + i0] * (1.0f - w)
           + an_bias[ha * AGENT + i1] * w + at_bias[e];
  }
  for (int e = tid; e < NH * T * AGENT; e += 256) {       // [h][t][a] flat == ta_bias flat
    int a = e % AGENT;
    int t = (e / AGENT) % T;
    int h = e / (T * AGENT);
    float coord = (t + 0.5f) * ((float)AGENT / (float)T) - 0.5f;
    coord = fminf(fmaxf(coord, 0.0f), (float)(AGENT - 1));
    int i0 = (int)floorf(coord);
    int i1 = min(i0 + 1, AGENT - 1);
    float w = coord - (float)i0;
    int base = (h * AGENT + a) * AGENT;
    sTA[e] = na_bias[base + i0] * (1.0f - w)
           + na_bias[base + i1] * w + ta_bias[e];
  }
  __syncthreads();

  // ---- P1: QKV GEMM (64x128 @ 128x384) with V_WMMA_F32_16X16X4_F32 ----
  // Wave w owns M-tile (w>>1); handles 12 N-tiles starting at (w&1)*12.
  {
    const int m0   = (wv >> 1) * 16;
    const int arow = m0 + lane_lo;
    v2f areg[32];
#pragma unroll
    for (int ks = 0; ks < 32; ++ks)
      areg[ks] = *(const v2f*)(sX + arow * XLD + ks * 4 + 2 * lane_hi);
#pragma unroll
    for (int nt = 0; nt < 12; ++nt) {
      const int n0 = ((wv & 1) * 12 + nt) * 16;
      const float* Wrow = w_qkv + (size_t)(n0 + lane_lo) * DIM + 2 * lane_hi;
      v2f breg[32];
#pragma unroll
      for (int ks = 0; ks < 32; ++ks)
        breg[ks] = *(const v2f*)(Wrow + ks * 4);
      v8f acc = {};
#pragma unroll
      for (int ks = 0; ks < 32; ++ks)
        acc = __builtin_amdgcn_wmma_f32_16x16x4_f32(false, areg[ks], false, breg[ks],
                                                    (short)0, acc, false, false);
      const int col = n0 + lane_lo;
      const float bias = b_qkv[col];
#pragma unroll
      for (int r = 0; r < 8; ++r)
        sQKV[(m0 + r + 8 * lane_hi) * QLD + col] = acc[r] + bias;
    }
  }
  __syncthreads();

  // ---- P2: depthwise conv1 (k=9, pad 4) over q[:,1:,:]  -> sHO[t][c] ----
  for (int e = tid; e < L * DIM; e += 256) {
    int t = e >> 7, c = e & 127;
    float acc = conv1_b[c];
    const float* wc = conv1_w + c * KSIZE;
#pragma unroll
    for (int j = 0; j < KSIZE; ++j) {
      int tt = t - (KSIZE / 2) + j;
      float qv = (tt >= 0 && tt < L) ? sQKV[(1 + tt) * QLD + c] : 0.0f;
      acc += wc[j] * qv;
    }
    sHO[t * OLD + c] = acc;
  }
  __syncthreads();

  // ---- P3: LayerNorm over channels per t, then exact GELU ----
  for (int t = wv; t < L; t += 8) {
    float vls[4]; float s = 0.0f;
#pragma unroll
    for (int i = 0; i < 4; ++i) { vls[i] = sHO[t * OLD + lane + 32 * i]; s += vls[i]; }
    float mu = wave_sum(s) * (1.0f / 128.0f);
    float s2 = 0.0f;
#pragma unroll
    for (int i = 0; i < 4; ++i) { float d = vls[i] - mu; s2 += d * d; }
    float var  = wave_sum(s2) * (1.0f / 128.0f);
    float rstd = rsqrtf(var + 1e-5f);
#pragma unroll
    for (int i = 0; i < 4; ++i) {
      int c = lane + 32 * i;
      float xn = (vls[i] - mu) * rstd * ln_w[c] + ln_b[c];
      sHO[t * OLD + c] = 0.5f * xn * (1.0f + erff(xn * 0.70710678118654752f));
    }
  }
  __syncthreads();

  // ---- P4: conv2 (stride 12) + tanh -> agent positions ----
  if (wv < AGENT) {
    int a = wv;
    float p = 0.0f;
    for (int idx = lane; idx < DIM * AGENT; idx += 32) {
      int c = idx / AGENT, j = idx % AGENT;
      p += conv2_w[c * AGENT + j] * sHO[(a * ASTRIDE + j) * OLD + c];
    }
    p = wave_sum(p);
    if (lane == 0) {
      const float refv[AGENT] = {0.0f, 14.75f, 29.5f, 44.25f, 59.0f};
      int pos = (int)(tanhf(p) + refv[a]);       // C truncation == jnp astype(int32)
      sPos[a] = min(max(pos, 0), L - 1);
    }
  }
  __syncthreads();

  // ---- P5: reference's flat-reshape gather + agent_q projection ----
  // xs.flat[i] = tok.flat[(i/5)*60 + pos[i%5]],  tok.flat over (60,128) = sX rows 1..60
  for (int o = tid; o < AGENT * DIM; o += 256) {
    int a2 = o >> 7, n = o & 127;
    float acc = b_projq[n];
    const float* wr = w_projq + (size_t)n * DIM;
    int i  = a2 * DIM;
    int dq = i / 5, r5 = i % 5;
    for (int dd = 0; dd < DIM; ++dd) {
      int j = dq * L + sPos[r5];
      acc += sX[(1 + (j >> 7)) * XLD + (j & 127)] * wr[dd];
      if (++r5 == AGENT) { r5 = 0; ++dq; }
    }
    sAgentQ[(n >> 4) * (AGENT * HD) + a2 * HD + (n & 15)] = acc;
  }
  __syncthreads();

  // ---- P6: attn1 scores (agent_q . k) + bias, in place; softmax over t ----
  for (int e = tid; e < NH * AGENT * T; e += 256) {
    int t = e % T;
    int a = (e / T) % AGENT;
    int h = e / (AGENT * T);
    const float* aq = sAgentQ + (h * AGENT + a) * HD;
    const float* kv = sQKV + t * QLD + DIM + h * HD;
    float d = 0.0f;
#pragma unroll
    for (int i = 0; i < HD; ++i) d += aq[i] * kv[i];
    sAT[e] += d * SCALEF;
  }
  __syncthreads();
  for (int r = wv; r < NH * AGENT; r += 8) {
    float* row = sAT + r * T;
    float v0 = row[lane];
    float v1 = (lane + 32 < T) ? row[lane + 32] : -3.4e38f;
    float m  = wave_max(fmaxf(v0, v1));
    float e0 = expf(v0 - m);
    float e1 = (lane + 32 < T) ? expf(v1 - m) : 0.0f;
    float inv = 1.0f / wave_sum(e0 + e1);
    row[lane] = e0 * inv;
    if (lane + 32 < T) row[lane + 32] = e1 * inv;
  }
  __syncthreads();

  // ---- P7: agent_v = attn1 @ v ; attn2 scores (q . agent_q) + bias; softmax over a ----
  for (int o = tid; o < NH * AGENT * HD; o += 256) {
    int d = o % HD;
    int a = (o / HD) % AGENT;
    int h = o / (HD * AGENT);
    const float* at = sAT + (h * AGENT + a) * T;
    const float* vv = sQKV + 2 * DIM + h * HD + d;
    float acc = 0.0f;
    for (int t = 0; t < T; ++t) acc += at[t] * vv[t * QLD];
    sAgentV[o] = acc;
  }
  for (int e = tid; e < NH * T * AGENT; e += 256) {
    int a = e % AGENT;
    int t = (e / AGENT) % T;
    int h = e / (T * AGENT);
    const float* qv = sQKV + t * QLD + h * HD;
    const float* aq = sAgentQ + (h * AGENT + a) * HD;
    float d = 0.0f;
#pragma unroll
    for (int i = 0; i < HD; ++i) d += qv[i] * aq[i];
    sTA[e] += d * SCALEF;
  }
  __syncthreads();
  for (int r = tid; r < NH * T; r += 256) {
    float* row = sTA + r * AGENT;
    float m = row[0];
#pragma unroll
    for (int i = 1; i < AGENT; ++i) m = fmaxf(m, row[i]);
    float ev[AGENT]; float s = 0.0f;
#pragma unroll
    for (int i = 0; i < AGENT; ++i) { ev[i] = expf(row[i] - m); s += ev[i]; }
    float inv = 1.0f / s;
#pragma unroll
    for (int i = 0; i < AGENT; ++i) row[i] = ev[i] * inv;
  }
  __syncthreads();

  // ---- P8: attn output = attn2 @ agent_v -> sHO rows 0..60 (61..63 still zero) ----
  for (int e = tid; e < T * DIM; e += 256) {
    int t = e >> 7, c = e & 127;
    int h = c >> 4, d = c & 15;
    const float* ta = sTA + (h * T + t) * AGENT;
    const float* av = sAgentV + h * (AGENT * HD) + d;
    float acc = 0.0f;
#pragma unroll
    for (int a = 0; a < AGENT; ++a) acc += ta[a] * av[a * HD];
    sHO[t * OLD + c] = acc;
  }
  __syncthreads();

  // ---- P9: final projection (64x128 @ 128x128) with WMMA -> global ----
  {
    float* ob = out + (size_t)b * (T * DIM);
    const int m0   = (wv >> 1) * 16;
    const int arow = m0 + lane_lo;
    v2f areg[32];
#pragma unroll
    for (int ks = 0; ks < 32; ++ks)
      areg[ks] = *(const v2f*)(sHO + arow * OLD + ks * 4 + 2 * lane_hi);
#pragma unroll
    for (int nt = 0; nt < 4; ++nt) {
      const int n0 = ((wv & 1) * 4 + nt) * 16;
      const float* Wrow = w_proj + (size_t)(n0 + lane_lo) * DIM + 2 * lane_hi;
      v2f breg[32];
#pragma unroll
      for (int ks = 0; ks < 32; ++ks)
        breg[ks] = *(const v2f*)(Wrow + ks * 4);
      v8f acc = {};
#pragma unroll
      for (int ks = 0; ks < 32; ++ks)
        acc = __builtin_amdgcn_wmma_f32_16x16x4_f32(false, areg[ks], false, breg[ks],
                                                    (short)0, acc, false, false);
      const int col = n0 + lane_lo;
      const float bias = b_proj[col];
#pragma unroll
      for (int r = 0; r < 8; ++r) {
        const int row = m0 + r + 8 * lane_hi;
        if (row < T) ob[row * DIM + col] = acc[r] + bias;
      }
    }
  }
}

extern "C" void kernel_launch(void* const* d_in, const int* in_sizes, int n_in,
                              void* d_out, int out_size, void* d_ws, size_t ws_size,
                              hipStream_t stream) {
  const float* x       = (const float*)d_in[0];
  const float* w_qkv   = (const float*)d_in[1];
  const float* b_qkv   = (const float*)d_in[2];
  const float* w_proj  = (const float*)d_in[3];
  const float* b_proj  = (const float*)d_in[4];
  const float* w_projq = (const float*)d_in[5];
  const float* b_projq = (const float*)d_in[6];
  const float* conv1_w = (const float*)d_in[7];
  const float* conv1_b = (const float*)d_in[8];
  const float* ln_w    = (const float*)d_in[9];
  const float* ln_b    = (const float*)d_in[10];
  const float* conv2_w = (const float*)d_in[11];
  const float* an_bias = (const float*)d_in[12];
  const float* na_bias = (const float*)d_in[13];
  const float* at_bias = (const float*)d_in[14];
  const float* ta_bias = (const float*)d_in[15];

  int num = in_sizes[0] / (T * DIM);
  agent_attn_fused<<<num, 256, 0, stream>>>(
      x, w_qkv, b_qkv, w_proj, b_proj, w_projq, b_projq,
      conv1_w, conv1_b, ln_w, ln_b, conv2_w,
      an_bias, na_bias, at_bias, ta_bias, (float*)d_out);
}